// TrivialAugmentWide_24232205484077
// MI455X (gfx1250) — compile-verified
//
#include <hip/hip_runtime.h>
#include <hip/hip_bf16.h>
#include <math.h>
#include <stdint.h>

// TrivialAugmentWide for MI455X (gfx1250).
// Memory-bound: ~77MB traffic => ~3.3us floor at 23.3 TB/s. No matmul => no
// WMMA; exercise CDNA5 memory paths instead: async global->LDS staging
// (ASYNCcnt + s_wait_asynccnt), LDS atomics for histograms, 128-bit loads,
// non-temporal 128-bit stores, global prefetch. All wave32-clean.

#define NB 64
#define NH 224
#define NW 224
#define NPIX (NH * NW)            // 50176
#define IMG_STRIDE (NPIX * 3)     // 150528 floats per image
#define CHUNK_F 3072              // floats per staged chunk (1024 pixels)
#define NCHUNK (IMG_STRIDE / CHUNK_F)  // 49, exact

// ---- gfx1250 async-to-LDS builtins (existence proven by round-1 probe) ----
#ifdef __HIP_DEVICE_COMPILE__
#ifdef __has_builtin
#if __has_builtin(__builtin_amdgcn_global_load_async_to_lds_b128) && \
    __has_builtin(__builtin_amdgcn_s_wait_asynccnt)
#define USE_ASYNC_LDS 1
#endif
#endif
#endif
#ifndef USE_ASYNC_LDS
#define USE_ASYNC_LDS 0
#endif

typedef int v4i __attribute__((ext_vector_type(4)));
typedef float v4f __attribute__((ext_vector_type(4)));

// ws float layout:
//   [0, NB)                  contrast "deg" per image
//   [NB, NB+6*NB)            autocontrast lo[3],hi[3] per image
//   [NB+6*NB, +NB*768)       equalize lut[3][256] per image
#define WS_DEG(b)  (b)
#define WS_LOHI(b) (NB + (b) * 6)
#define WS_LUT(b)  (NB + 6 * NB + (b) * 768)

__device__ __forceinline__ float u8f(float x) {
  return floorf(fminf(fmaxf(x, 0.0f), 255.0f));
}
__device__ __forceinline__ float grayf(float r, float g, float b) {
  // jnp.round = round-half-even; device default rounding is RTNE -> rintf.
  return rintf(r * 0.2989f + g * 0.5870f + b * 0.1140f);
}
__device__ __forceinline__ unsigned umn(unsigned a, unsigned b) { return a < b ? a : b; }
__device__ __forceinline__ unsigned umx(unsigned a, unsigned b) { return a > b ? a : b; }

// accumulate stats over 1024 staged pixels
__device__ __forceinline__ void taw_accum(int op, const float* buf, int t,
                                          unsigned& gacc,
                                          unsigned& l0, unsigned& l1, unsigned& l2,
                                          unsigned& h0, unsigned& h1, unsigned& h2,
                                          unsigned* shist) {
  for (int p = t; p < 1024; p += 256) {
    float u0 = u8f(buf[p * 3 + 0]);
    float u1 = u8f(buf[p * 3 + 1]);
    float u2 = u8f(buf[p * 3 + 2]);
    if (op == 8) {                       // contrast: sum of rounded luma
      gacc += (unsigned)grayf(u0, u1, u2);
    } else if (op == 12) {               // autocontrast: per-channel min/max
      unsigned a = (unsigned)u0, b = (unsigned)u1, c = (unsigned)u2;
      l0 = umn(l0, a); h0 = umx(h0, a);
      l1 = umn(l1, b); h1 = umx(h1, b);
      l2 = umn(l2, c); h2 = umx(h2, c);
    } else {                             // equalize: 3x256 LDS histogram
      atomicAdd(&shist[(unsigned)u0], 1u);
      atomicAdd(&shist[256u + (unsigned)u1], 1u);
      atomicAdd(&shist[512u + (unsigned)u2], 1u);
    }
  }
}

extern "C" __global__ __launch_bounds__(256)
void taw_stats(const float* __restrict__ img, const int* __restrict__ opi,
               float* __restrict__ ws) {
  const int b = blockIdx.x;
  const int op = opi[b];
  if (op != 8 && op != 12 && op != 13) return;  // block-uniform early out
  const int t = threadIdx.x;
  const float* __restrict__ src = img + (size_t)b * IMG_STRIDE;

  __shared__ __align__(16) float stage[2][CHUNK_F];  // 24KB double buffer
  __shared__ unsigned shist[768];
  __shared__ unsigned sred[256];
  __shared__ unsigned sminmax[6];
  __shared__ unsigned slastc;

  if (op == 13) {
    for (int i = t; i < 768; i += 256) shist[i] = 0u;
  }
  if (op == 12 && t < 6) sminmax[t] = (t < 3) ? 255u : 0u;
  unsigned gacc = 0u;
  unsigned l0 = 255u, l1 = 255u, l2 = 255u, h0 = 0u, h1 = 0u, h2 = 0u;
  __syncthreads();

#if USE_ASYNC_LDS
  // ROCm clang-22 types these builtins by payload width: b128 -> int4*.
#define TAW_ASYNC_B128(GP, LP) \
  __builtin_amdgcn_global_load_async_to_lds_b128((v4i*)(GP), (v4i*)(LP), 0, 0)
  {  // prologue: stage chunk 0 (3 x b128 per lane = 4KB per wave round)
    const float* gp = src + t * 4;
    float* lp = &stage[0][t * 4];
    TAW_ASYNC_B128(gp, lp);
    TAW_ASYNC_B128(gp + 1024, lp + 1024);
    TAW_ASYNC_B128(gp + 2048, lp + 2048);
  }
  for (int k = 0; k < NCHUNK; k++) {
    if (k + 1 < NCHUNK) {  // stage next chunk into the other buffer
      const float* gp = src + (k + 1) * CHUNK_F + t * 4;
      float* lp = &stage[(k + 1) & 1][t * 4];
      TAW_ASYNC_B128(gp, lp);
      TAW_ASYNC_B128(gp + 1024, lp + 1024);
      TAW_ASYNC_B128(gp + 2048, lp + 2048);
      __builtin_amdgcn_s_wait_asynccnt(3);  // chunk k's 3 ops complete
    } else {
      __builtin_amdgcn_s_wait_asynccnt(0);
    }
    __syncthreads();  // all waves' async data visible
    taw_accum(op, stage[k & 1], t, gacc, l0, l1, l2, h0, h1, h2, shist);
    __syncthreads();  // done reading before buffer is restaged
  }
#undef TAW_ASYNC_B128
#else
  for (int k = 0; k < NCHUNK; k++) {
    const v4f* gp = (const v4f*)(src + k * CHUNK_F);
    v4f* lp = (v4f*)stage[0];
    lp[t] = gp[t];
    lp[t + 256] = gp[t + 256];
    lp[t + 512] = gp[t + 512];
    __syncthreads();
    taw_accum(op, stage[0], t, gacc, l0, l1, l2, h0, h1, h2, shist);
    __syncthreads();
  }
#endif

  if (op == 8) {
    sred[t] = gacc;
    __syncthreads();
    for (int off = 128; off > 0; off >>= 1) {
      if (t < off) sred[t] += sred[t + off];
      __syncthreads();
    }
    // 50176*255 < 2^24 -> fp32 sum exact; mean = sum/n then floor
    if (t == 0) ws[WS_DEG(b)] = floorf((float)sred[0] / 50176.0f);
  } else if (op == 12) {
    atomicMin(&sminmax[0], l0); atomicMin(&sminmax[1], l1); atomicMin(&sminmax[2], l2);
    atomicMax(&sminmax[3], h0); atomicMax(&sminmax[4], h1); atomicMax(&sminmax[5], h2);
    __syncthreads();
    if (t < 6) ws[WS_LOHI(b) + t] = (float)sminmax[t];
  } else {  // equalize: per-channel last-nonzero, cumsum scan, LUT
    for (int c = 0; c < 3; c++) {
      if (t == 0) slastc = 0u;
      __syncthreads();
      unsigned hv = shist[c * 256 + t];
      if (hv) atomicMax(&slastc, (unsigned)t);
      sred[t] = hv;
      __syncthreads();
      for (int off = 1; off < 256; off <<= 1) {  // Hillis-Steele inclusive scan
        unsigned add = (t >= off) ? sred[t - off] : 0u;
        __syncthreads();
        sred[t] += add;
        __syncthreads();
      }
      unsigned last = slastc;
      unsigned step = (50176u - shist[c * 256 + last]) / 255u;
      float lv;
      if (step == 0u) {
        lv = (float)t;  // identity: where(step==0, chi, ...)
      } else if (t == 0) {
        lv = 0.0f;      // lut shifted right with 0 prepended
      } else {
        unsigned q = (sred[t - 1] + step / 2u) / step;
        lv = (float)(q > 255u ? 255u : q);
      }
      ws[WS_LUT(b) + c * 256 + t] = lv;
      __syncthreads();
    }
  }
}

// 4 pixels per thread: 12 consecutive floats = 3 aligned b128 loads/NT stores.
extern "C" __global__ __launch_bounds__(256)
void taw_apply(const float* __restrict__ img, const int* __restrict__ opi,
               const int* __restrict__ bini, const int* __restrict__ sgn,
               const float* __restrict__ ws, float* __restrict__ out) {
  const int b = blockIdx.y;
  const int p0 = (blockIdx.x * 256 + threadIdx.x) * 4;  // 49*256*4 == 50176
  const int op = opi[b];
  const float* __restrict__ src = img + (size_t)b * IMG_STRIDE;
  float* __restrict__ dst = out + (size_t)b * IMG_STRIDE + (size_t)p0 * 3;

  // magnitude table evaluated in fp64 like np.linspace, cast to fp32
  const double di = (double)bini[b];
  float mag;
  switch (op) {
    case 1: case 2: case 6: case 7: case 8: case 9:
      mag = (float)(di * (0.99 / 30.0)); break;
    case 3: case 4:
      mag = (float)(di * (32.0 / 30.0)); break;
    case 5:
      mag = (float)(di * (135.0 / 30.0)); break;
    case 10:
      mag = (float)(8.0 - rint(di * (1.0 / 5.0))); break;  // no .5 ties occur
    case 11:
      mag = (float)(255.0 - 8.5 * di); break;
    default:
      mag = 0.0f; break;
  }
  const bool issigned = (op >= 1 && op <= 8);
  const float sg = issigned ? (sgn[b] > 0 ? 1.0f : -1.0f) : 1.0f;
  const float m = mag * sg;

  __builtin_prefetch(src + (size_t)p0 * 3 + 12288, 0, 0);  // global_prefetch_b8

  float o[12];
  if (op >= 1 && op <= 5) {
    // ---- affine (shear/translate/rotate), bilinear, zero fill ----
    float m0, m1, m2, m3, m4, m5;
    if (op == 1)      { m0 = 1.f; m1 = m;   m2 = 0.f; m3 = 0.f; m4 = 1.f; m5 = 0.f; }
    else if (op == 2) { m0 = 1.f; m1 = 0.f; m2 = 0.f; m3 = m;   m4 = 1.f; m5 = 0.f; }
    else if (op == 3) { m0 = 1.f; m1 = 0.f; m2 = m;   m3 = 0.f; m4 = 1.f; m5 = 0.f; }
    else if (op == 4) { m0 = 1.f; m1 = 0.f; m2 = 0.f; m3 = 0.f; m4 = 1.f; m5 = m;   }
    else {
      float rr = (-3.14159265358979323846f * m) / 180.0f;
      float cs = cosf(rr), sn = sinf(rr);
      const float cx = 112.0f, cy = 112.0f;
      m0 = cs; m1 = -sn; m2 = (1.0f - cs) * cx + sn * cy;
      m3 = sn; m4 = cs;  m5 = -sn * cx + (1.0f - cs) * cy;
    }
#pragma unroll
    for (int q = 0; q < 4; q++) {
      const int pix = p0 + q;
      const int y = pix / NW;
      const int x = pix - y * NW;
      float xin = m0 * (float)x + m1 * (float)y + m2;
      float yin = m3 * (float)x + m4 * (float)y + m5;
      float x0 = floorf(xin), y0 = floorf(yin);
      float wx = xin - x0, wy = yin - y0;
      float a0 = 0.f, a1 = 0.f, a2 = 0.f;
#pragma unroll
      for (int dy = 0; dy < 2; dy++) {
#pragma unroll
        for (int dx = 0; dx < 2; dx++) {
          float xc = x0 + (float)dx, yc = y0 + (float)dy;
          bool valid = (xc >= 0.f) && (xc <= (float)(NW - 1)) &&
                       (yc >= 0.f) && (yc <= (float)(NH - 1));
          if (valid) {
            int xi = (int)xc, yi = (int)yc;
            const float* sp = src + ((size_t)yi * NW + xi) * 3;
            float w = (dx ? wx : (1.0f - wx)) * (dy ? wy : (1.0f - wy));
            a0 += u8f(sp[0]) * w;
            a1 += u8f(sp[1]) * w;
            a2 += u8f(sp[2]) * w;
          }
        }
      }
      o[q * 3 + 0] = u8f(a0);
      o[q * 3 + 1] = u8f(a1);
      o[q * 3 + 2] = u8f(a2);
    }
  } else if (op == 9) {
    // ---- sharpness: 3x3 depthwise smooth on interior, blend ----
    const float f = m + 1.0f;
    const float w1 = 1.0f / 13.0f, w5 = 5.0f / 13.0f;
#pragma unroll
    for (int q = 0; q < 4; q++) {
      const int pix = p0 + q;
      const int y = pix / NW;
      const int x = pix - y * NW;
      const float* cp = src + (size_t)pix * 3;
      float u0 = u8f(cp[0]), u1 = u8f(cp[1]), u2 = u8f(cp[2]);
      if (x == 0 || x == NW - 1 || y == 0 || y == NH - 1) {
        o[q * 3 + 0] = u0; o[q * 3 + 1] = u1; o[q * 3 + 2] = u2;
      } else {
        float d0 = 0.f, d1 = 0.f, d2 = 0.f;
#pragma unroll
        for (int dy = -1; dy <= 1; dy++) {
#pragma unroll
          for (int dx = -1; dx <= 1; dx++) {
            float w = (dx == 0 && dy == 0) ? w5 : w1;
            const float* sp = src + ((size_t)(y + dy) * NW + (x + dx)) * 3;
            d0 += u8f(sp[0]) * w;
            d1 += u8f(sp[1]) * w;
            d2 += u8f(sp[2]) * w;
          }
        }
        float r0 = u8f(d0), r1 = u8f(d1), r2 = u8f(d2);
        o[q * 3 + 0] = u8f(r0 + f * (u0 - r0));
        o[q * 3 + 1] = u8f(r1 + f * (u1 - r1));
        o[q * 3 + 2] = u8f(r2 + f * (u2 - r2));
      }
    }
  } else {
    // ---- pure per-pixel ops: 3 coalesced b128 loads ----
    v4f A = *(const v4f*)(src + (size_t)p0 * 3);
    v4f Bv = *(const v4f*)(src + (size_t)p0 * 3 + 4);
    v4f Cv = *(const v4f*)(src + (size_t)p0 * 3 + 8);
    float u[12] = {u8f(A.x),  u8f(A.y),  u8f(A.z),  u8f(A.w),
                   u8f(Bv.x), u8f(Bv.y), u8f(Bv.z), u8f(Bv.w),
                   u8f(Cv.x), u8f(Cv.y), u8f(Cv.z), u8f(Cv.w)};
    switch (op) {
      case 6: {  // brightness: blend(0, u, m+1)
        float f = m + 1.0f;
#pragma unroll
        for (int i = 0; i < 12; i++) o[i] = u8f(f * u[i]);
      } break;
      case 7: {  // color: blend(gray, u, m+1)
        float f = m + 1.0f;
#pragma unroll
        for (int q = 0; q < 4; q++) {
          float g = grayf(u[q * 3], u[q * 3 + 1], u[q * 3 + 2]);
#pragma unroll
          for (int c = 0; c < 3; c++)
            o[q * 3 + c] = u8f(g + f * (u[q * 3 + c] - g));
        }
      } break;
      case 8: {  // contrast: blend(deg, u, m+1), deg from stats pass
        float f = m + 1.0f;
        float dg = ws[WS_DEG(b)];
#pragma unroll
        for (int i = 0; i < 12; i++) o[i] = u8f(dg + f * (u[i] - dg));
      } break;
      case 10: {  // posterize
        int sh = 8 - (int)m;
        sh = sh < 0 ? 0 : (sh > 8 ? 8 : sh);
#pragma unroll
        for (int i = 0; i < 12; i++) o[i] = (float)((((int)u[i]) >> sh) << sh);
      } break;
      case 11: {  // solarize
        int th = (int)m;
#pragma unroll
        for (int i = 0; i < 12; i++) {
          int a = (int)u[i];
          o[i] = (float)(a >= th ? 255 - a : a);
        }
      } break;
      case 12: {  // autocontrast using stats-pass lo/hi
        const float* lh = ws + WS_LOHI(b);
#pragma unroll
        for (int c = 0; c < 3; c++) {
          float lo = lh[c], hi = lh[3 + c];
          float sc = 255.0f / (hi - lo);
#pragma unroll
          for (int q = 0; q < 4; q++) {
            int i = q * 3 + c;
            o[i] = (hi > lo) ? u8f(u[i] * sc - lo * sc) : u[i];
          }
        }
      } break;
      case 13: {  // equalize via stats-pass LUT
        const float* lut = ws + WS_LUT(b);
#pragma unroll
        for (int q = 0; q < 4; q++)
#pragma unroll
          for (int c = 0; c < 3; c++)
            o[q * 3 + c] = lut[c * 256 + (int)u[q * 3 + c]];
      } break;
      default: {  // identity
#pragma unroll
        for (int i = 0; i < 12; i++) o[i] = u[i];
      } break;
    }
  }
  // write-once output stream -> 3 non-temporal b128 stores per thread
  v4f s0 = {o[0], o[1], o[2], o[3]};
  v4f s1 = {o[4], o[5], o[6], o[7]};
  v4f s2 = {o[8], o[9], o[10], o[11]};
  __builtin_nontemporal_store(s0, (v4f*)(dst));
  __builtin_nontemporal_store(s1, (v4f*)(dst + 4));
  __builtin_nontemporal_store(s2, (v4f*)(dst + 8));
}

extern "C" void kernel_launch(void* const* d_in, const int* in_sizes, int n_in,
                              void* d_out, int out_size, void* d_ws, size_t ws_size,
                              hipStream_t stream) {
  (void)in_sizes; (void)n_in; (void)out_size; (void)ws_size;
  const float* img = (const float*)d_in[0];
  const int* opi = (const int*)d_in[1];
  const int* bini = (const int*)d_in[2];
  const int* sgn = (const int*)d_in[3];
  float* out = (float*)d_out;
  float* ws = (float*)d_ws;

  taw_stats<<<NB, 256, 0, stream>>>(img, opi, ws);
  dim3 grid(NPIX / (256 * 4), NB);  // 49 x 64
  taw_apply<<<grid, 256, 0, stream>>>(img, opi, bini, sgn, ws, out);
}